// GCNNet_77008763617440
// MI455X (gfx1250) — compile-verified
//
#include <hip/hip_runtime.h>
#include <hip/hip_bf16.h>

typedef __attribute__((ext_vector_type(2))) float v2f;
typedef __attribute__((ext_vector_type(8))) float v8f;

// ---------------------------------------------------------------------------
// WMMA f32 GEMM: C[M x NCOL] = A[M x K] * B[K x NCOL] (+ bias), row-major.
// One wave -> one 16x16 tile, V_WMMA_F32_16X16X4_F32 over K in steps of 4.
// 4 waves / block cover 64 rows. K, NCOL compile-time so the k-loop unrolls.
// ---------------------------------------------------------------------------
template <int K, int NCOL>
__launch_bounds__(128)
__global__ void wmma_gemm_f32(const float* __restrict__ A,
                              const float* __restrict__ B,
                              float* __restrict__ C,
                              const float* __restrict__ bias,
                              int M) {
  const int wave = threadIdx.x >> 5;
  const int lane = threadIdx.x & 31;
  const int g    = lane >> 4;    // lane-half selects K pair (A) / K pair (B)
  const int ln   = lane & 15;    // m for A, n for B/C
  const int row0 = (blockIdx.x * 4 + wave) * 16;
  const int col0 = blockIdx.y * 16;
  if (row0 >= M) return;         // uniform per wave -> EXEC stays all-ones

  const float* __restrict__ arow = A + (size_t)(row0 + ln) * K + 2 * g;
  const float* __restrict__ bcol = B + (size_t)(2 * g) * NCOL + (col0 + ln);

  v8f acc = {};
#pragma unroll
  for (int k = 0; k < K; k += 4) {
    v2f a = *(const v2f*)(arow + k);          // A[m][k+2g .. k+2g+1]
    v2f b;
    b.x = bcol[(size_t)k * NCOL];             // B[k+2g  ][n]
    b.y = bcol[(size_t)(k + 1) * NCOL];       // B[k+2g+1][n]
    acc = __builtin_amdgcn_wmma_f32_16x16x4_f32(
        /*neg_a=*/false, a, /*neg_b=*/false, b,
        /*c_mod=*/(short)0, acc, /*reuse_a=*/false, /*reuse_b=*/false);
  }

  const float bv = bias ? bias[col0 + ln] : 0.0f;
  float* crow = C + (size_t)(row0 + 8 * g) * NCOL + (col0 + ln);
#pragma unroll
  for (int r = 0; r < 8; ++r)                 // D vgpr r -> row (r + 8g)
    crow[(size_t)r * NCOL] = acc[r] + bv;
}

// ---------------------------------------------------------------------------
// Normalization helpers
// ---------------------------------------------------------------------------
__global__ void fill_f32(float* __restrict__ p, float v, long long n) {
  long long i = blockIdx.x * (long long)blockDim.x + threadIdx.x;
  if (i < n) p[i] = v;
}

__global__ void deg_count(const int* __restrict__ ei, float* __restrict__ deg,
                          int E) {
  int e = blockIdx.x * blockDim.x + threadIdx.x;
  if (e < E) atomicAdd(&deg[ei[E + e]], 1.0f);   // dst = ei[1][e]
}

__global__ void rsqrt_inplace(float* __restrict__ p, int n) {
  int i = blockIdx.x * blockDim.x + threadIdx.x;
  if (i < n) p[i] = rsqrtf(p[i]);                // deg >= 1 (self loop)
}

// ---------------------------------------------------------------------------
// Edge scatter: agg[dst] += dis[src]*dis[dst] * H[src]; 1 lane = 4 channels.
// agg buffers (25.6 MB) are L2-resident, so float atomics resolve in L2.
// ---------------------------------------------------------------------------
__global__ void edge_scatter(const int* __restrict__ ei,
                             const float* __restrict__ H,
                             const float* __restrict__ dis,
                             float* __restrict__ agg, int E) {
  const int NCH = 128;
  long long tid = blockIdx.x * (long long)blockDim.x + threadIdx.x;
  long long e = tid >> 5;                 // 32 lanes per edge
  int c = (int)(tid & 31) * 4;
  if (e >= E) return;
  int s = ei[e];
  int d = ei[E + e];
  float w = dis[s] * dis[d];
  const float4 hv = *(const float4*)(H + (size_t)s * NCH + c);
  float* out = agg + (size_t)d * NCH + c;
  atomicAdd(out + 0, w * hv.x);
  atomicAdd(out + 1, w * hv.y);
  atomicAdd(out + 2, w * hv.z);
  atomicAdd(out + 3, w * hv.w);
}

// ---------------------------------------------------------------------------
// h = relu(agg + dis^2 * HW + b)   (self-loop term added analytically)
// out may alias agg (pure element-wise).
// ---------------------------------------------------------------------------
__global__ void finalize_relu(const float* agg, const float* __restrict__ HW,
                              const float* __restrict__ dis,
                              const float* __restrict__ b, float* out, int n) {
  const int NCH = 128;
  long long idx = blockIdx.x * (long long)blockDim.x + threadIdx.x;
  if (idx >= (long long)n * NCH) return;
  int row = (int)(idx >> 7);
  int c = (int)(idx & 127);
  float di = dis[row];
  float v = agg[idx] + di * di * HW[idx] + b[c];
  out[idx] = v > 0.0f ? v : 0.0f;
}

// ---------------------------------------------------------------------------
extern "C" void kernel_launch(void* const* d_in, const int* in_sizes, int n_in,
                              void* d_out, int out_size, void* d_ws,
                              size_t ws_size, hipStream_t stream) {
  const float* x  = (const float*)d_in[0];
  const int*   ei = (const int*)d_in[1];
  const float* W1 = (const float*)d_in[2];
  const float* b1 = (const float*)d_in[3];
  const float* W2 = (const float*)d_in[4];
  const float* b2 = (const float*)d_in[5];
  const float* Wd = (const float*)d_in[6];
  const float* bd = (const float*)d_in[7];

  const int NIN = 256, NHID = 128, NOUT = 64;
  const int N = in_sizes[0] / NIN;   // 50000
  const int E = in_sizes[1] / 2;     // 500000

  float* out = (float*)d_out;                    // [N, 64]
  float* h2  = out + (size_t)N * NOUT;           // [N, 128] (2nd output)

  float* dis  = (float*)d_ws;                    // [N]
  float* bufA = dis + N;                         // [N, 128] GEMM outputs
  float* bufB = bufA + (size_t)N * NHID;         // [N, 128] agg / h1

  const long long nfeat = (long long)N * NHID;
  dim3 blk(256);
  auto cdiv = [](long long a, long long b) { return (unsigned)((a + b - 1) / b); };

  // --- symmetric normalization: dis = rsqrt(1 + in_degree) ---
  fill_f32<<<cdiv(N, 256), blk, 0, stream>>>(dis, 1.0f, N);
  deg_count<<<cdiv(E, 256), blk, 0, stream>>>(ei, dis, E);
  rsqrt_inplace<<<cdiv(N, 256), blk, 0, stream>>>(dis, N);

  dim3 gblk(128);
  dim3 g1(cdiv(N, 64), NHID / 16);
  dim3 g3(cdiv(N, 64), NOUT / 16);

  // --- layer 1: bufA = x @ W1; bufB = scatter + self + b1, relu -> h1 ---
  wmma_gemm_f32<256, 128><<<g1, gblk, 0, stream>>>(x, W1, bufA, nullptr, N);
  fill_f32<<<cdiv(nfeat, 256), blk, 0, stream>>>(bufB, 0.0f, nfeat);
  edge_scatter<<<cdiv((long long)E * 32, 256), blk, 0, stream>>>(ei, bufA, dis, bufB, E);
  finalize_relu<<<cdiv(nfeat, 256), blk, 0, stream>>>(bufB, bufA, dis, b1, bufB, N);

  // --- layer 2: bufA = h1 @ W2; h2 (in d_out) = relu(scatter + self + b2) ---
  wmma_gemm_f32<128, 128><<<g1, gblk, 0, stream>>>(bufB, W2, bufA, nullptr, N);
  fill_f32<<<cdiv(nfeat, 256), blk, 0, stream>>>(bufB, 0.0f, nfeat);
  edge_scatter<<<cdiv((long long)E * 32, 256), blk, 0, stream>>>(ei, bufA, dis, bufB, E);
  finalize_relu<<<cdiv(nfeat, 256), blk, 0, stream>>>(bufB, bufA, dis, b2, h2, N);

  // --- decoder: out = h2 @ Wd + bd ---
  wmma_gemm_f32<128, 64><<<g3, gblk, 0, stream>>>(h2, Wd, out, bd, N);
}